// CustomEncoder_35923106464234
// MI455X (gfx1250) — compile-verified
//
#include <hip/hip_runtime.h>
#include <math.h>

// MI455X / gfx1250. Reference: 2-layer graph conv.
//   denom = att.sum(axis=0)                       (400MB stream -> HBM bound, ~17us floor)
//   h  = relu( scatter(x@W1 * alpha) + b1 )
//   out =      scatter(h@W2 * alpha) + b2
// alpha(e) = exp(att[src,dst] - denom[dst])
// GEMMs are tiny (1.3 GFLOP) -> fp32 WMMA (V_WMMA_F32_16X16X4_F32) keeps
// reference precision while using the matrix pipe. The att stream uses
// non-temporal b128 loads so it doesn't evict the L2-resident h/agg/out
// matrices that the scatter passes re-read.

typedef __attribute__((ext_vector_type(2))) float v2f;
typedef __attribute__((ext_vector_type(4))) float v4f;
typedef __attribute__((ext_vector_type(8))) float v8f;

// ---------------------------------------------------------------- fill zero
__global__ void fill_zero_kernel(float* __restrict__ p, int n) {
  int i = blockIdx.x * blockDim.x + threadIdx.x;
  if (i < n) p[i] = 0.0f;
}

// ------------------------------------------------- out[i] = b[i & (C-1)]
__global__ void fill_bias_kernel(float* __restrict__ out,
                                 const float* __restrict__ b,
                                 int n, int cmask) {
  int i = blockIdx.x * blockDim.x + threadIdx.x;
  if (i < n) out[i] = b[i & cmask];
}

// ------------------------------------ denom[col..col+3] += sum_r att[r][..]
// b128 non-temporal loads: 4 columns per thread, coalesced across the wave,
// grid.y tiles row chunks, partial sums combined with f32 atomics.
__global__ void colsum4_kernel(const float* __restrict__ att,
                               float* __restrict__ denom,
                               int N, int N4, int rowsPerChunk) {
  int c4 = blockIdx.x * blockDim.x + threadIdx.x;
  if (c4 >= N4) return;
  long r0 = (long)blockIdx.y * rowsPerChunk;
  long r1 = r0 + rowsPerChunk;
  if (r1 > N) r1 = N;
  v4f acc = {0.0f, 0.0f, 0.0f, 0.0f};
  const v4f* row = (const v4f*)att + r0 * N4 + c4;
  for (long r = r0; r < r1; ++r, row += N4) {
    v4f v = __builtin_nontemporal_load(row);   // NT: don't pollute L2
    acc += v;
  }
  float* d = denom + c4 * 4;
  atomicAdd(d + 0, acc.x);
  atomicAdd(d + 1, acc.y);
  atomicAdd(d + 2, acc.z);
  atomicAdd(d + 3, acc.w);
}

// scalar fallback (only used if N % 4 != 0)
__global__ void colsum_kernel(const float* __restrict__ att,
                              float* __restrict__ denom,
                              int N, int rowsPerChunk) {
  int col = blockIdx.x * blockDim.x + threadIdx.x;
  if (col >= N) return;
  long r0 = (long)blockIdx.y * rowsPerChunk;
  long r1 = r0 + rowsPerChunk;
  if (r1 > N) r1 = N;
  float acc = 0.0f;
  for (long r = r0; r < r1; ++r) acc += att[r * N + col];
  atomicAdd(&denom[col], acc);
}

// --------------------------------------------------------- fp32 WMMA GEMM
// C[M x Ncol] = A[M x K] @ B[K x Ncol], row-major, M%16==0, K%4==0, Ncol%16==0.
// 4 waves per block; each wave owns one 16x16 output tile and iterates
// V_WMMA_F32_16X16X4_F32 over K.
//
// Lane layouts (CDNA5 ISA 7.12.2, wave32):
//   A 16x4 : lanes 0-15 -> M=lane, v0=K0,v1=K1 ; lanes 16-31 -> v0=K2,v1=K3
//   B 4x16 : lanes 0-15 -> N=lane, v0=K0,v1=K1 ; lanes 16-31 -> v0=K2,v1=K3
//   C 16x16: VGPR r, lanes 0-15 -> (M=r,   N=lane),
//                    lanes16-31 -> (M=r+8, N=lane-16)
__global__ __launch_bounds__(128) void gemm_wmma_f32(
    const float* __restrict__ A, const float* __restrict__ B,
    float* __restrict__ C, int M, int K, int Ncol) {
  const int lane = threadIdx.x & 31;
  const int wave = threadIdx.x >> 5;
  const int colBase = (blockIdx.x * 4 + wave) * 16;
  const int rowBase = blockIdx.y * 16;
  if (colBase >= Ncol) return;          // uniform per-wave: EXEC stays all-ones

  const int half = lane >> 4;           // 0: K0/K1 side, 1: K2/K3 side
  const int l15  = lane & 15;
  const int aRow = rowBase + l15;       // A row owned by this lane
  const int kOff = half * 2;
  const int bCol = colBase + l15;       // B/C column owned by this lane

  v8f acc = {};
  for (int k = 0; k < K; k += 4) {
    v2f a, b;
    const float* ap = A + (long)aRow * K + (k + kOff);
    a.x = ap[0];                        // consecutive K -> single b64 load
    a.y = ap[1];
    b.x = B[(long)(k + kOff)     * Ncol + bCol];
    b.y = B[(long)(k + kOff + 1) * Ncol + bCol];
    acc = __builtin_amdgcn_wmma_f32_16x16x4_f32(
        /*neg_a=*/false, a, /*neg_b=*/false, b,
        /*c_mod=*/(short)0, acc, /*reuse_a=*/false, /*reuse_b=*/false);
  }

#pragma unroll
  for (int r = 0; r < 8; ++r) {
    int row = rowBase + r + half * 8;
    C[(long)row * Ncol + bCol] = acc[r];
  }
}

// -------------------------------------------------------------- edge scatter
// agg[dst] += h[src] * exp(att[src,dst] - denom[dst])
// Each thread: 4 channels of one edge (float4 gather from L2-resident h,
// broadcast load of the shared att element, 4 f32 atomics into L2).
__global__ __launch_bounds__(256) void scatter_kernel(
    const float* __restrict__ h, const int* __restrict__ src,
    const int* __restrict__ dst, const float* __restrict__ att,
    const float* __restrict__ denom, float* __restrict__ agg,
    int nEdges, int N, int C) {
  const int gpe = C >> 2;                             // thread-groups per edge
  const int epb = blockDim.x / gpe;                   // edges per block
  const int e = blockIdx.x * epb + (int)(threadIdx.x / gpe);
  if (e >= nEdges) return;
  const int c = (threadIdx.x % gpe) * 4;
  const int s = src[e];
  const int d = dst[e];
  const float alpha = expf(att[(long)s * N + d] - denom[d]);
  const float4 hv = *(const float4*)(h + (long)s * C + c);
  float* o = agg + (long)d * C + c;
  atomicAdd(o + 0, hv.x * alpha);
  atomicAdd(o + 1, hv.y * alpha);
  atomicAdd(o + 2, hv.z * alpha);
  atomicAdd(o + 3, hv.w * alpha);
}

// ----------------------------------------------------------------- ReLU
__global__ void relu_kernel(float* __restrict__ p, int n) {
  int i = blockIdx.x * blockDim.x + threadIdx.x;
  if (i < n) p[i] = fmaxf(p[i], 0.0f);
}

// ================================================================ launcher
extern "C" void kernel_launch(void* const* d_in, const int* in_sizes, int n_in,
                              void* d_out, int out_size, void* d_ws, size_t ws_size,
                              hipStream_t stream) {
  const float* x   = (const float*)d_in[0];
  const int*   ei  = (const int*)  d_in[1];
  const float* att = (const float*)d_in[2];
  const float* W1  = (const float*)d_in[3];
  const float* b1  = (const float*)d_in[4];
  const float* W2  = (const float*)d_in[5];
  const float* b2  = (const float*)d_in[6];
  float* out = (float*)d_out;

  const int IN_C  = 128;
  const int N     = in_sizes[0] / IN_C;        // 10000 (625 * 16)
  const int HID   = in_sizes[3] / IN_C;        // 256
  const int OUT_C = in_sizes[5] / HID;         // 128
  const int nE    = in_sizes[1] / 2;           // 330000
  const int* src = ei;
  const int* dst = ei + nE;

  // workspace layout: denom[N] | h[N*HID] | agg[N*HID]
  char* ws = (char*)d_ws;
  size_t off = ((size_t)N * sizeof(float) + 255) & ~(size_t)255;
  float* denom = (float*)ws;
  float* h     = (float*)(ws + off);
  float* agg   = h + (size_t)N * HID;

  // 1) denom = att column sums (HBM-bound 400MB NT stream)
  fill_zero_kernel<<<(N + 255) / 256, 256, 0, stream>>>(denom, N);
  if ((N & 3) == 0) {
    const int N4 = N / 4;
    const int rowsPerChunk = 128;
    dim3 g((N4 + 255) / 256, (N + rowsPerChunk - 1) / rowsPerChunk);
    colsum4_kernel<<<g, 256, 0, stream>>>(att, denom, N, N4, rowsPerChunk);
  } else {
    const int rowsPerChunk = 256;
    dim3 g((N + 255) / 256, (N + rowsPerChunk - 1) / rowsPerChunk);
    colsum_kernel<<<g, 256, 0, stream>>>(att, denom, N, rowsPerChunk);
  }

  // 2) h = x @ W1   (fp32 WMMA)
  {
    dim3 g(HID / 64, N / 16);
    gemm_wmma_f32<<<g, 128, 0, stream>>>(x, W1, h, N, IN_C, HID);
  }

  // 3) agg = b1 ; agg[dst] += h[src]*alpha ; agg = relu(agg)
  fill_bias_kernel<<<((N * HID) + 255) / 256, 256, 0, stream>>>(agg, b1, N * HID, HID - 1);
  {
    const int epb = 256 / (HID / 4);             // edges per 256-thread block
    scatter_kernel<<<(nE + epb - 1) / epb, 256, 0, stream>>>(
        h, src, dst, att, denom, agg, nE, N, HID);
  }
  relu_kernel<<<((N * HID) + 255) / 256, 256, 0, stream>>>(agg, N * HID);

  // 4) h = agg @ W2  (fp32 WMMA, reuse h buffer)
  {
    dim3 g(OUT_C / 64, N / 16);
    gemm_wmma_f32<<<g, 128, 0, stream>>>(agg, W2, h, N, HID, OUT_C);
  }

  // 5) out = b2 ; out[dst] += h[src]*alpha
  fill_bias_kernel<<<((N * OUT_C) + 255) / 256, 256, 0, stream>>>(out, b2, N * OUT_C, OUT_C - 1);
  {
    const int epb = 256 / (OUT_C / 4);
    scatter_kernel<<<(nE + epb - 1) / epb, 256, 0, stream>>>(
        h, src, dst, att, denom, out, nE, N, OUT_C);
  }
}